// DistanceFromAnswerLoss_86921548137027
// MI455X (gfx1250) — compile-verified
//
#include <hip/hip_runtime.h>

#define BATCH 4096
#define C_LEN 8192
#define COEFF 0.1f

typedef float v8f __attribute__((ext_vector_type(8)));
typedef float v2f __attribute__((ext_vector_type(2)));

// Reduce 64 per-wave partial sums (2 per lane) to a single float using the
// f32 matrix pipe: D = A(16x4) x ones(4x16). Every D[m][n] = rowsum_m, so
// summing the 8 D VGPRs gives rows 0-7 on lanes 0-15 and rows 8-15 on lanes
// 16-31; two shuffles finish the reduction. EXEC must be all-ones here
// (callers keep this call unconditional for all 256 threads).
__device__ __forceinline__ float wave_reduce_wmma(float a0, float a1) {
    v2f A; A[0] = a0; A[1] = a1;
    v2f B; B[0] = 1.0f; B[1] = 1.0f;
    v8f C = {};
    v8f D = __builtin_amdgcn_wmma_f32_16x16x4_f32(
        /*neg_a=*/false, A, /*neg_b=*/false, B,
        /*c_mod=*/(short)0, C, /*reuse_a=*/false, /*reuse_b=*/false);
    float s = D[0] + D[1] + D[2] + D[3] + D[4] + D[5] + D[6] + D[7];
    float lo = __shfl(s, 0, 32);   // rows 0-7
    float hi = __shfl(s, 16, 32);  // rows 8-15
    return lo + hi;
}

__global__ void __launch_bounds__(256)
row_loss_kernel(const float* __restrict__ input,
                const int* __restrict__ target,
                float* __restrict__ block_sums) {
    const int row = blockIdx.x;
    const int tid = threadIdx.x;

    const long long t = (long long)target[row];
    const float tf = (float)t;
    const float* __restrict__ rowp = input + (size_t)row * C_LEN;

    // 8192 elements / 256 threads = 32 per thread = 8 x float4 (coalesced).
    float acc0 = 0.0f, acc1 = 0.0f;
#pragma unroll
    for (int k = 0; k < 8; ++k) {
        const int base = tid * 4 + k * 1024;
        const float4 v = *reinterpret_cast<const float4*>(rowp + base);
        const float d0 = (float)base - tf;
        acc0 += fabsf(d0)        * __expf(v.x);
        acc1 += fabsf(d0 + 1.0f) * __expf(v.y);
        acc0 += fabsf(d0 + 2.0f) * __expf(v.z);
        acc1 += fabsf(d0 + 3.0f) * __expf(v.w);
    }

    // Closed-form row norm: sum_{i=0}^{C-1}(i-t)^2 = S(t) + S(C-1-t),
    // S(n) = n(n+1)(2n+1)/6. Exact in int64 for C = 8192.
    const long long r  = (long long)(C_LEN - 1) - t;
    const long long ss = t * (t + 1) * (2 * t + 1) / 6
                       + r * (r + 1) * (2 * r + 1) / 6;
    const float scale = (t == 0) ? 0.0f : (float)(1.0 / sqrt((double)ss));
    acc0 *= scale;
    acc1 *= scale;

    // Unconditional for the whole block: EXEC all-ones for WMMA.
    const float wave_sum = wave_reduce_wmma(acc0, acc1);

    __shared__ float wsum[8];
    const int lane = tid & 31;
    const int wid  = tid >> 5;
    if (lane == 0) wsum[wid] = wave_sum;
    __syncthreads();
    if (tid == 0) {
        float s = 0.0f;
#pragma unroll
        for (int i = 0; i < 8; ++i) s += wsum[i];
        block_sums[row] = s;
    }
}

// Deterministic final reduction: fixed summation order, no float atomics.
__global__ void __launch_bounds__(256)
final_reduce_kernel(const float* __restrict__ block_sums,
                    float* __restrict__ out) {
    __shared__ float sm[256];
    const int tid = threadIdx.x;
    float s = 0.0f;
#pragma unroll
    for (int k = 0; k < BATCH / 256; ++k) s += block_sums[tid + k * 256];
    sm[tid] = s;
    __syncthreads();
    for (int off = 128; off > 0; off >>= 1) {
        if (tid < off) sm[tid] += sm[tid + off];
        __syncthreads();
    }
    if (tid == 0) out[0] = COEFF * sm[0];
}

extern "C" void kernel_launch(void* const* d_in, const int* in_sizes, int n_in,
                              void* d_out, int out_size, void* d_ws, size_t ws_size,
                              hipStream_t stream) {
    (void)in_sizes; (void)n_in; (void)out_size; (void)ws_size;
    const float* input  = (const float*)d_in[0];
    const int*   target = (const int*)d_in[1];
    float*       out    = (float*)d_out;
    float*       bsums  = (float*)d_ws;  // 4096 floats = 16 KiB scratch

    row_loss_kernel<<<BATCH, 256, 0, stream>>>(input, target, bsums);
    final_reduce_kernel<<<1, 256, 0, stream>>>(bsums, out);
}